// GIN_5l_2826088481299
// MI455X (gfx1250) — compile-verified
//
#include <hip/hip_runtime.h>
#include <hip/hip_bf16.h>

#define NN      50000
#define NE      1250000
#define DIMF    64
#define NG      128
#define OUTC    10
#define BN_EPS  1e-5f

typedef __attribute__((ext_vector_type(2))) float v2f;
typedef __attribute__((ext_vector_type(8))) float v8f;

// ---------------------------------------------------------------------------
// init: ones vector (for plain-bias GEMM epilogue) + zero the pooling buffer
// ---------------------------------------------------------------------------
__global__ void gin_init_misc(float* ones, float* gbuf) {
    int t = blockIdx.x * blockDim.x + threadIdx.x;
    if (t < DIMF) ones[t] = 1.0f;
    if (t < NG * DIMF) gbuf[t] = 0.0f;
}

// ---------------------------------------------------------------------------
// fold BatchNorm(eval) + bias into per-channel scale/shift:
//   z = (y + b1 - rmean) * gamma * rsqrt(var+eps) + beta = y*s + sh
// ---------------------------------------------------------------------------
__global__ void gin_bn_prep(const float* __restrict__ b1,
                            const float* __restrict__ gamma,
                            const float* __restrict__ beta,
                            const float* __restrict__ rmean,
                            const float* __restrict__ rvar,
                            float* __restrict__ scale,
                            float* __restrict__ shift) {
    int c = threadIdx.x;
    float s = gamma[c] * rsqrtf(rvar[c] + BN_EPS);
    scale[c] = s;
    shift[c] = (b1[c] - rmean[c]) * s + beta[c];
}

// ---------------------------------------------------------------------------
// agg = h (copy, float4 vectorized); edge atomics add the neighborhood sum
// ---------------------------------------------------------------------------
__global__ void gin_copy_f4(const float4* __restrict__ in, float4* __restrict__ out) {
    int i = blockIdx.x * blockDim.x + threadIdx.x;   // NN*DIMF/4 = 800k threads
    out[i] = in[i];
}

__global__ void gin_edge_scatter(const float* __restrict__ h,
                                 const int* __restrict__ src,
                                 const int* __restrict__ dst,
                                 float* __restrict__ agg) {
    int gid = blockIdx.x * 256 + threadIdx.x;        // NE*64 = 80M threads
    int e = gid >> 6;
    int c = gid & 63;
    int s = src[e];
    int d = dst[e];
    atomicAdd(&agg[(size_t)d * DIMF + c], h[(size_t)s * DIMF + c]);
}

// ---------------------------------------------------------------------------
// O[r, n] = relu( (A[r, :] @ W[:, n]) * scale[n] + shift[n] )
// A: [NN x 64] row-major, W: [64 x 64] row-major.
// One block = 128 threads = 4 waves; block covers 16 rows x 64 cols,
// each wave owns one 16x16 tile, K-loop of 16 x V_WMMA_F32_16X16X4_F32.
// 50000 rows = 3125 blocks exactly.
// ---------------------------------------------------------------------------
__global__ void __launch_bounds__(128)
gin_gemm64_affine_relu(const float* __restrict__ A,
                       const float* __restrict__ W,
                       const float* __restrict__ scale,
                       const float* __restrict__ shift,
                       float* __restrict__ O) {
    const int wave   = threadIdx.x >> 5;       // 0..3 -> N tile
    const int lane   = threadIdx.x & 31;
    const int mrow   = lane & 15;              // row within 16-row tile
    const int khalf  = (lane >> 4) << 1;       // 0 or 2 (K sub-offset)
    const int ncol   = (wave << 4) + (lane & 15);
    const int rowb   = blockIdx.x << 4;

    const float* __restrict__ arow = A + (size_t)(rowb + mrow) * DIMF;

    v8f acc = {};
    #pragma unroll
    for (int k = 0; k < DIMF; k += 4) {
        v2f a, b;
        // A fragment: lane holds A[row, k+khalf] and A[row, k+khalf+1] (8B load)
        a.x = arow[k + khalf];
        a.y = arow[k + khalf + 1];
        // B fragment: lane holds W[k+khalf, ncol] and W[k+khalf+1, ncol]
        b.x = W[(k + khalf) * DIMF + ncol];
        b.y = W[(k + khalf + 1) * DIMF + ncol];
        acc = __builtin_amdgcn_wmma_f32_16x16x4_f32(
            /*neg_a=*/false, a, /*neg_b=*/false, b,
            /*c_mod=*/(short)0, acc, /*reuse_a=*/false, /*reuse_b=*/false);
    }

    const float s  = scale[ncol];
    const float sh = shift[ncol];
    const int   rb = rowb + ((lane >> 4) ? 8 : 0);   // upper half-wave -> M+8
    #pragma unroll
    for (int r = 0; r < 8; ++r) {
        float v = acc[r] * s + sh;
        v = v > 0.0f ? v : 0.0f;
        O[(size_t)(rb + r) * DIMF + ncol] = v;
    }
}

// ---------------------------------------------------------------------------
// global_add_pool: atomic scatter of node features into per-graph rows
// ---------------------------------------------------------------------------
__global__ void gin_pool(const float* __restrict__ h,
                         const int* __restrict__ batch,
                         float* __restrict__ gbuf) {
    int gid = blockIdx.x * 256 + threadIdx.x;        // NN*64 = 3.2M threads
    int node = gid >> 6;
    int c = gid & 63;
    atomicAdd(&gbuf[(size_t)batch[node] * DIMF + c], h[(size_t)node * DIMF + c]);
}

// ---------------------------------------------------------------------------
// head: hid = relu(g @ lin1_w + lin1_b)   (128 x 64, trivial)
// ---------------------------------------------------------------------------
__global__ void gin_head1(const float* __restrict__ gbuf,
                          const float* __restrict__ W,   // [64 x 64]
                          const float* __restrict__ b,
                          float* __restrict__ hid) {
    int g = blockIdx.x;
    int c = threadIdx.x;
    const float* row = gbuf + (size_t)g * DIMF;
    float s = b[c];
    for (int k = 0; k < DIMF; ++k) s += row[k] * W[k * DIMF + c];
    hid[(size_t)g * DIMF + c] = fmaxf(s, 0.0f);
}

// ---------------------------------------------------------------------------
// logits = hid @ lin2_w + lin2_b ; out = log_softmax(logits)
// ---------------------------------------------------------------------------
__global__ void gin_head2(const float* __restrict__ hid,
                          const float* __restrict__ W,   // [64 x 10]
                          const float* __restrict__ b,
                          float* __restrict__ out) {
    __shared__ float logits[OUTC];
    __shared__ float lse;
    int g = blockIdx.x;
    int t = threadIdx.x;
    if (t < OUTC) {
        float s = b[t];
        const float* row = hid + (size_t)g * DIMF;
        for (int k = 0; k < DIMF; ++k) s += row[k] * W[k * OUTC + t];
        logits[t] = s;
    }
    __syncthreads();
    if (t == 0) {
        float m = logits[0];
        for (int i = 1; i < OUTC; ++i) m = fmaxf(m, logits[i]);
        float se = 0.0f;
        for (int i = 0; i < OUTC; ++i) se += __expf(logits[i] - m);
        lse = m + __logf(se);
    }
    __syncthreads();
    if (t < OUTC) out[(size_t)g * OUTC + t] = logits[t] - lse;
}

// ---------------------------------------------------------------------------
// launch
// ---------------------------------------------------------------------------
extern "C" void kernel_launch(void* const* d_in, const int* in_sizes, int n_in,
                              void* d_out, int out_size, void* d_ws, size_t ws_size,
                              hipStream_t stream) {
    (void)in_sizes; (void)n_in; (void)out_size; (void)ws_size;

    const float* x     = (const float*)d_in[0];
    const int*   eidx  = (const int*)d_in[1];
    const int*   src   = eidx;
    const int*   dst   = eidx + NE;
    const int*   batch = (const int*)d_in[2];
    // convs: per layer i, base 3+8i: w1,b1,gamma,beta,rmean,rvar,w2,b2
    const float* lin1_w = (const float*)d_in[43];
    const float* lin1_b = (const float*)d_in[44];
    const float* lin2_w = (const float*)d_in[45];
    const float* lin2_b = (const float*)d_in[46];

    float* ws   = (float*)d_ws;
    float* U    = ws;                         // [NN x 64]
    float* V    = U + (size_t)NN * DIMF;      // [NN x 64]
    float* Z    = V + (size_t)NN * DIMF;      // [NN x 64]
    float* gbuf = Z + (size_t)NN * DIMF;      // [128 x 64]
    float* hid  = gbuf + NG * DIMF;           // [128 x 64]
    float* scl  = hid + NG * DIMF;            // [5 x 64]
    float* shf  = scl + 5 * DIMF;             // [5 x 64]
    float* ones = shf + 5 * DIMF;             // [64]

    gin_init_misc<<<(NG * DIMF + 255) / 256, 256, 0, stream>>>(ones, gbuf);

    for (int i = 0; i < 5; ++i) {
        const float* b1    = (const float*)d_in[3 + 8 * i + 1];
        const float* gamma = (const float*)d_in[3 + 8 * i + 2];
        const float* beta  = (const float*)d_in[3 + 8 * i + 3];
        const float* rmean = (const float*)d_in[3 + 8 * i + 4];
        const float* rvar  = (const float*)d_in[3 + 8 * i + 5];
        gin_bn_prep<<<1, DIMF, 0, stream>>>(b1, gamma, beta, rmean, rvar,
                                            scl + i * DIMF, shf + i * DIMF);
    }

    const float* h = x;
    for (int i = 0; i < 5; ++i) {
        const float* w1 = (const float*)d_in[3 + 8 * i + 0];
        const float* w2 = (const float*)d_in[3 + 8 * i + 6];
        const float* b2 = (const float*)d_in[3 + 8 * i + 7];
        float* agg = (i & 1) ? V : U;

        // agg = h + segment_sum(h[src], dst)
        gin_copy_f4<<<(NN * DIMF / 4) / 256, 256, 0, stream>>>(
            (const float4*)h, (float4*)agg);
        gin_edge_scatter<<<(NE * 64) / 256, 256, 0, stream>>>(h, src, dst, agg);

        // z = relu(bn(agg @ w1 + b1));  h' = relu(z @ w2 + b2)
        gin_gemm64_affine_relu<<<NN / 16, 128, 0, stream>>>(
            agg, w1, scl + i * DIMF, shf + i * DIMF, Z);
        gin_gemm64_affine_relu<<<NN / 16, 128, 0, stream>>>(
            Z, w2, ones, b2, agg);
        h = agg;
    }

    gin_pool<<<(NN * DIMF) / 256, 256, 0, stream>>>(h, batch, gbuf);
    gin_head1<<<NG, DIMF, 0, stream>>>(gbuf, lin1_w, lin1_b, hid);
    gin_head2<<<NG, 32, 0, stream>>>(hid, lin2_w, lin2_b, (float*)d_out);
}